// GCN_50062138802536
// MI455X (gfx1250) — compile-verified
//
#include <hip/hip_runtime.h>
#include <hip/hip_bf16.h>
#include <math.h>

typedef __attribute__((ext_vector_type(2))) float v2f;
typedef __attribute__((ext_vector_type(8))) float v8f;

#define NNODES 65536
#define NEDGES 1048576
#define NGRAPH 4096

// ---------------------------------------------------------------- utilities
__global__ void fill_f32(float* __restrict__ p, float v, int n) {
    int i = blockIdx.x * blockDim.x + threadIdx.x;
    if (i < n) p[i] = v;
}

__global__ void accum_deg(const int* __restrict__ dst, float* __restrict__ deg, int E) {
    int e = blockIdx.x * blockDim.x + threadIdx.x;
    if (e < E) atomicAdd(&deg[dst[e]], 1.0f);
}

__global__ void deg_to_dinv(float* __restrict__ d, int n) {
    int i = blockIdx.x * blockDim.x + threadIdx.x;
    if (i < n) d[i] = rsqrtf(d[i]);
}

// Pad rows of X [n,kin] -> Xp [n,kout] with zeros (kout >= kin)
__global__ void pad_rows(const float* __restrict__ X, float* __restrict__ Xp,
                         int n, int kin, int kout) {
    int idx = blockIdx.x * blockDim.x + threadIdx.x;
    if (idx >= n * kout) return;
    int row = idx / kout;
    int c   = idx - row * kout;
    Xp[idx] = (c < kin) ? X[row * kin + c] : 0.0f;
}

// Pack B [Korig,NCorig] into WMMA B-fragment order, zero-padded:
// Bp[((nt*KS + kk)*32 + lane)*2 + j] = B[kk*4 + (lane>>4)*2 + j][nt*16 + (lane&15)]
__global__ void pack_b(const float* __restrict__ B, float* __restrict__ Bp,
                       int Korig, int NCorig, int Kp, int NT) {
    int KS = Kp >> 2;
    int total = NT * KS * 64;
    int idx = blockIdx.x * blockDim.x + threadIdx.x;
    if (idx >= total) return;
    int j    = idx & 1;
    int lane = (idx >> 1) & 31;
    int rest = idx >> 6;
    int kk   = rest % KS;
    int nt   = rest / KS;
    int ka   = (kk << 2) + ((lane >> 4) << 1) + j;
    int col  = (nt << 4) + (lane & 15);
    Bp[idx] = (ka < Korig && col < NCorig) ? B[ka * NCorig + col] : 0.0f;
}

// ---------------------------------------------------- fp32 WMMA tiled GEMM
// C[M,NC] = A[M,K] @ B (fragment-packed) (+bias, +relu). lda = K (padded,
// K%4==0), ldc = NC. One wave computes a 64x16 slab (4 M-tiles), branch-free
// inner loop of V_WMMA_F32_16X16X4_F32 with b64 vector operand loads.
template<int K, int NC>
__global__ __launch_bounds__(256)
void wmma_gemm(const float* __restrict__ A, const float* __restrict__ Bpack,
               const float* __restrict__ bias, float* __restrict__ C,
               int M, int relu)
{
    constexpr int MT  = 4;               // M-tiles per wave
    constexpr int NCP = (NC + 15) & ~15;
    constexpr int NT  = NCP >> 4;        // N tiles
    constexpr int KS  = K >> 2;          // K steps of 4

    int wave = (blockIdx.x * blockDim.x + threadIdx.x) >> 5;  // wave-uniform
    int lane = threadIdx.x & 31;
    int mwaves = M / (16 * MT);
    if (wave >= mwaves * NT) return;                          // uniform exit
    int mw = wave / NT;
    int nt = wave - mw * NT;
    int row0 = mw * 16 * MT;

    int lhalf = lane >> 4;
    int l16   = lane & 15;
    int col   = (nt << 4) + l16;

    // A: lane covers row (row0 + m*16 + l16), K-pair offset lhalf*2 per step
    const float* abase = A + (size_t)(row0 + l16) * K + (lhalf << 1);
    // B: coalesced packed fragments for this n-tile
    const v2f* bp = (const v2f*)Bpack + (size_t)nt * KS * 32 + lane;

    v8f acc[MT];
    #pragma unroll
    for (int m = 0; m < MT; ++m) acc[m] = (v8f)0.0f;

    #pragma unroll 5
    for (int kk = 0; kk < KS; ++kk) {
        v2f b = bp[kk * 32];
        #pragma unroll
        for (int m = 0; m < MT; ++m) {
            v2f a = *(const v2f*)(abase + (size_t)m * 16 * K + kk * 4);
            acc[m] = __builtin_amdgcn_wmma_f32_16x16x4_f32(
                false, a, false, b, (short)0, acc[m], false, false);
        }
    }

    if (col < NC) {
        float bv = bias ? bias[col] : 0.0f;
        #pragma unroll
        for (int m = 0; m < MT; ++m) {
            #pragma unroll
            for (int i = 0; i < 8; ++i) {
                int row = row0 + m * 16 + i + (lhalf << 3);
                float v = acc[m][i] + bv;
                if (relu) v = fmaxf(v, 0.0f);
                C[(size_t)row * NC + col] = v;
            }
        }
    }
}

// --------------------------------------------- edge gather/scatter (L2 hot)
__global__ void edge_aggregate(const int* __restrict__ src, const int* __restrict__ dst,
                               const float* __restrict__ dinv, const float* __restrict__ hw,
                               float* __restrict__ agg, int E, int F)
{
    int gid  = blockIdx.x * blockDim.x + threadIdx.x;
    int e    = gid >> 5;
    int lane = gid & 31;
    if (e >= E) return;
    int s = src[e], d = dst[e];
    float coef = dinv[s] * dinv[d];
    const float* hs = hw  + (size_t)s * F;
    float*       ad = agg + (size_t)d * F;
    for (int f = lane; f < F; f += 32)
        atomicAdd(&ad[f], coef * hs[f]);
}

// out = relu(agg + hw * dinv^2 + bias)
__global__ void combine_relu(const float* __restrict__ agg, const float* __restrict__ hw,
                             const float* __restrict__ dinv, const float* __restrict__ bias,
                             float* __restrict__ out, int n, int F)
{
    int idx = blockIdx.x * blockDim.x + threadIdx.x;
    if (idx >= n * F) return;
    int node = idx / F;
    int f    = idx - node * F;
    float di = dinv[node];
    float v = agg[idx] + hw[idx] * di * di + bias[f];
    out[idx] = fmaxf(v, 0.0f);
}

// per-graph product of 16 nodes per channel, via sign-parity * exp(sum log|x|)
__global__ void mul_aggregate(const float* __restrict__ h, float* __restrict__ p, int Gn)
{
    int idx = blockIdx.x * blockDim.x + threadIdx.x;
    if (idx >= Gn * 6) return;
    int g = idx / 6;
    int c = idx - g * 6;
    int neg = 0;
    float slog = 0.0f;
    #pragma unroll
    for (int i = 0; i < 16; ++i) {
        float v = h[(size_t)(g * 16 + i) * 6 + c];
        neg += (v < 0.0f) ? 1 : 0;
        slog += logf(fabsf(v));
    }
    float sign = (neg & 1) ? -1.0f : 1.0f;
    p[idx] = sign * expf(slog);
}

// softmax over dim 0 (4096 graphs) independently per channel (6 blocks)
__global__ void softmax_dim0(const float* __restrict__ p, float* __restrict__ out, int Gn)
{
    __shared__ float red[256];
    int c = blockIdx.x;
    int tid = threadIdx.x;
    float m = -INFINITY;
    for (int g = tid; g < Gn; g += 256) m = fmaxf(m, p[g * 6 + c]);
    red[tid] = m; __syncthreads();
    for (int s = 128; s > 0; s >>= 1) { if (tid < s) red[tid] = fmaxf(red[tid], red[tid + s]); __syncthreads(); }
    float mx = red[0]; __syncthreads();
    float sum = 0.0f;
    for (int g = tid; g < Gn; g += 256) sum += expf(p[g * 6 + c] - mx);
    red[tid] = sum; __syncthreads();
    for (int s = 128; s > 0; s >>= 1) { if (tid < s) red[tid] += red[tid + s]; __syncthreads(); }
    float tot = red[0];
    for (int g = tid; g < Gn; g += 256) out[g * 6 + c] = expf(p[g * 6 + c] - mx) / tot;
}

// ---------------------------------------------------------------- launcher
extern "C" void kernel_launch(void* const* d_in, const int* in_sizes, int n_in,
                              void* d_out, int out_size, void* d_ws, size_t ws_size,
                              hipStream_t stream)
{
    const float* x   = (const float*)d_in[0];   // [N,19]
    const float* W1  = (const float*)d_in[1];   // [19,100]
    const float* b1  = (const float*)d_in[2];   // [100]
    const float* Wc1 = (const float*)d_in[3];   // [100,100]
    const float* bc1 = (const float*)d_in[4];   // [100]
    const float* Wc2 = (const float*)d_in[5];   // [100,40]
    const float* bc2 = (const float*)d_in[6];   // [40]
    const float* W2  = (const float*)d_in[7];   // [40,6]
    const float* b2  = (const float*)d_in[8];   // [6]
    const int*   ei  = (const int*)d_in[9];     // [2,E]
    const int*   src = ei;
    const int*   dst = ei + NEDGES;
    float* out = (float*)d_out;                 // [4096,6]

    // workspace layout (floats)
    float* ws    = (float*)d_ws;
    float* dinv  = ws;                             // N
    float* apad  = dinv + NNODES;                  // N*20    padded x
    float* bpack = apad + (size_t)NNODES * 20;     // 16384   packed weights
    float* buf0  = bpack + 16384;                  // N*100   h1 / h2 / h3
    float* buf1  = buf0 + (size_t)NNODES * 100;    // N*100   hw1 / hw2 / h4
    float* buf2  = buf1 + (size_t)NNODES * 100;    // N*100   agg1 / agg2 / p

    const int T = 256;
    auto blocks = [](long n, int t) { return (int)((n + t - 1) / t); };

    // degrees with self loop -> dinv = rsqrt(deg)
    fill_f32<<<blocks(NNODES, T), T, 0, stream>>>(dinv, 1.0f, NNODES);
    accum_deg<<<blocks(NEDGES, T), T, 0, stream>>>(dst, dinv, NEDGES);
    deg_to_dinv<<<blocks(NNODES, T), T, 0, stream>>>(dinv, NNODES);

    // pad x to [N,20]
    pad_rows<<<blocks((long)NNODES * 20, T), T, 0, stream>>>(x, apad, NNODES, 19, 20);

    // h1 = relu(xpad @ W1 + b1)   [N,100]  (K=20 padded, Korig=19)
    pack_b<<<blocks(7L * 5 * 64, T), T, 0, stream>>>(W1, bpack, 19, 100, 20, 7);
    wmma_gemm<20, 100><<<blocks((65536L/64) * 7 * 32, T), T, 0, stream>>>(
        apad, bpack, b1, buf0, NNODES, 1);

    // conv1: hw1 = h1 @ Wc1; agg; h2 = relu(agg + hw1*dinv^2 + bc1)
    pack_b<<<blocks(7L * 25 * 64, T), T, 0, stream>>>(Wc1, bpack, 100, 100, 100, 7);
    wmma_gemm<100, 100><<<blocks((65536L/64) * 7 * 32, T), T, 0, stream>>>(
        buf0, bpack, nullptr, buf1, NNODES, 0);
    fill_f32<<<blocks((long)NNODES * 100, T), T, 0, stream>>>(buf2, 0.0f, NNODES * 100);
    edge_aggregate<<<blocks((long)NEDGES * 32, T), T, 0, stream>>>(src, dst, dinv, buf1, buf2, NEDGES, 100);
    combine_relu<<<blocks((long)NNODES * 100, T), T, 0, stream>>>(buf2, buf1, dinv, bc1, buf0, NNODES, 100);

    // conv2: hw2 = h2 @ Wc2; agg; h3 = relu(agg + hw2*dinv^2 + bc2)
    pack_b<<<blocks(3L * 25 * 64, T), T, 0, stream>>>(Wc2, bpack, 100, 40, 100, 3);
    wmma_gemm<100, 40><<<blocks((65536L/64) * 3 * 32, T), T, 0, stream>>>(
        buf0, bpack, nullptr, buf1, NNODES, 0);
    fill_f32<<<blocks((long)NNODES * 40, T), T, 0, stream>>>(buf2, 0.0f, NNODES * 40);
    edge_aggregate<<<blocks((long)NEDGES * 32, T), T, 0, stream>>>(src, dst, dinv, buf1, buf2, NEDGES, 40);
    combine_relu<<<blocks((long)NNODES * 40, T), T, 0, stream>>>(buf2, buf1, dinv, bc2, buf0, NNODES, 40);

    // h4 = h3 @ W2 + b2   [N,6]
    pack_b<<<blocks(1L * 10 * 64, T), T, 0, stream>>>(W2, bpack, 40, 6, 40, 1);
    wmma_gemm<40, 6><<<blocks((65536L/64) * 1 * 32, T), T, 0, stream>>>(
        buf0, bpack, b2, buf1, NNODES, 0);

    // per-graph product -> p [4096,6], softmax over graphs -> out
    mul_aggregate<<<blocks((long)NGRAPH * 6, T), T, 0, stream>>>(buf1, buf2, NGRAPH);
    softmax_dim0<<<6, 256, 0, stream>>>(buf2, out, NGRAPH);
}